// HybridEdgeConv_82617990906083
// MI455X (gfx1250) — compile-verified
//
#include <hip/hip_runtime.h>
#include <math.h>

typedef __attribute__((ext_vector_type(2))) float v2f;
typedef __attribute__((ext_vector_type(8))) float v8f;

#define BB 128
#define NN 256
#define KNB 16
__device__ __constant__ float kBN = 0.999995000037498f; // 1/sqrt(1+1e-5)

__device__ __forceinline__ v8f wmma_f32(v2f a, v2f b, v8f c) {
  // V_WMMA_F32_16X16X4_F32: D = A(16x4) * B(4x16) + C, fp32 exact
  return __builtin_amdgcn_wmma_f32_16x16x4_f32(
      /*neg_a=*/false, a, /*neg_b=*/false, b,
      /*c_mod=*/(short)0, c, /*reuse_a=*/false, /*reuse_b=*/false);
}

// ---------------------------------------------------------------- layer-1 pad
// Zero-pad (B,N,6) -> (B,N,8) so every kNN fragment load is an aligned b64
// with no bounds guards (zeros are Gram-neutral).
__global__ __launch_bounds__(256) void pad_kernel(const float* __restrict__ x,
                                                  float* __restrict__ xp) {
  int i = blockIdx.x * 256 + threadIdx.x;
  if (i >= BB * NN) return;
  const float* s = x + (size_t)i * 6;
  float* d = xp + (size_t)i * 8;
#pragma unroll
  for (int k = 0; k < 6; ++k) d[k] = s[k];
  d[6] = 0.f;
  d[7] = 0.f;
}

// ---------------------------------------------------------------- sq norms
template <int Din>
__global__ __launch_bounds__(256) void sq_kernel(const float* __restrict__ x,
                                                 float* __restrict__ sq) {
  int i = blockIdx.x * 256 + threadIdx.x;
  if (i >= BB * NN) return;
  const float* p = x + (size_t)i * Din;
  float s = 0.f;
#pragma unroll
  for (int d = 0; d < Din; ++d) s += p[d] * p[d];
  sq[i] = s;
}

// ---------------------------------------------------------------- fused kNN
// block = (batch b, 16-row tile it): 8 waves build d2[16][256] with fp32 WMMA
// Gram tiles, then per-row top-16 via wave32 shuffle min-extraction.
// Din must be a multiple of 4 (layer 1 uses the padded Din=8 buffer).
template <int Din>
__global__ __launch_bounds__(256) void knn_kernel(const float* __restrict__ x,
                                                  const float* __restrict__ sq,
                                                  int* __restrict__ idx) {
  static_assert((Din & 3) == 0, "pad Din to a multiple of 4");
  int b = blockIdx.x >> 4;
  int it = blockIdx.x & 15;
  int tid = threadIdx.x;
  int lane = tid & 31;
  int wave = tid >> 5;  // 0..7
  __shared__ float ds[16][NN];  // 16KB d2 stripe
  __shared__ float sqi[16];
  if (tid < 16) sqi[tid] = sq[b * NN + it * 16 + tid];
  __syncthreads();

  const float* xb = x + (size_t)b * NN * Din;
  int hi = lane >> 4;   // K offset +2 for upper half-wave
  int l15 = lane & 15;

  for (int jt = wave; jt < 16; jt += 8) {
    v8f acc = {};
    const float* arow = xb + (size_t)(it * 16 + l15) * Din;
    const float* brow = xb + (size_t)(jt * 16 + l15) * Din;
#pragma unroll 8
    for (int kk = 0; kk < Din; kk += 4) {
      int k0 = kk + hi * 2;
      v2f a = *(const v2f*)(arow + k0);   // global_load_b64, unconditional
      v2f w = *(const v2f*)(brow + k0);
      acc = wmma_f32(a, w, acc);
    }
    float sqj = sq[b * NN + jt * 16 + l15];
    int colbase = jt * 16 + l15;
    if (jt == it) {  // wave-uniform: only this tile holds diagonal entries
#pragma unroll
      for (int r = 0; r < 8; ++r) {
        int M = r + hi * 8;
        float d2 = sqi[M] + sqj - 2.0f * acc[r];
        if (M == l15) d2 = INFINITY;  // exclude self
        ds[M][colbase] = d2;
      }
    } else {
#pragma unroll
      for (int r = 0; r < 8; ++r) {
        int M = r + hi * 8;
        ds[M][colbase] = sqi[M] + sqj - 2.0f * acc[r];
      }
    }
  }
  __syncthreads();

  for (int rr = wave; rr < 16; rr += 8) {
    float v[8];
    int taken = 0;
#pragma unroll
    for (int s = 0; s < 8; ++s) v[s] = ds[rr][lane + 32 * s];
    int row = it * 16 + rr;
    for (int sel = 0; sel < KNB; ++sel) {
      float bv = INFINITY;
      int bc = 0x7fffffff;
#pragma unroll
      for (int s = 0; s < 8; ++s) {
        if (!(taken & (1 << s))) {
          float val = v[s];
          int c = lane + 32 * s;
          if (val < bv || (val == bv && c < bc)) { bv = val; bc = c; }
        }
      }
#pragma unroll
      for (int off = 16; off > 0; off >>= 1) {
        float ov = __shfl_xor(bv, off, 32);
        int oc = __shfl_xor(bc, off, 32);
        if (ov < bv || (ov == bv && oc < bc)) { bv = ov; bc = oc; }
      }
      if ((bc & 31) == lane) taken |= 1 << (bc >> 5);  // owning lane retires it
      if (lane == 0) idx[(b * NN + row) * KNB + sel] = bc;
    }
  }
}

// ---------------------------------------------------------------- EdgeConv
// block = one point (b,n), 8 waves. Stage A = [xi | xj-xi] (16 x 2Din, even
// padded stride -> ds_load_b64 fragments, conflict-free). Each wave owns
// 16-wide output-channel tiles; unrolled fp32 WMMA chain over K=2Din; fused
// bias + BN + ReLU + max-over-16-neighbors epilogue.
template <int Din, int Cout>
__global__ __launch_bounds__(256) void econv_kernel(
    const float* __restrict__ x, const int* __restrict__ idx,
    const float* __restrict__ W, const float* __restrict__ bias,
    const float* __restrict__ g, const float* __restrict__ be,
    float* __restrict__ out) {
  constexpr int K2 = 2 * Din;
  constexpr int As = K2 + 2;      // even stride: 8B-aligned pairs, stride%64==2
  constexpr int nct = Cout / 16;

  int p = blockIdx.x;
  int b = p >> 8;
  int n = p & 255;
  int tid = threadIdx.x;
  int lane = tid & 31, wave = tid >> 5;

  __shared__ float xi[Din];
  __shared__ float A[16 * As];
  __shared__ int nbr[KNB];

  const float* xb = x + (size_t)b * NN * Din;
  if (tid < Din) xi[tid] = xb[(size_t)n * Din + tid];
  if (tid < KNB) nbr[tid] = idx[(size_t)p * KNB + tid];
  __syncthreads();
  for (int t = tid; t < KNB * Din; t += 256) {
    int k = t / Din, d = t - k * Din;
    float xiv = xi[d];
    float xjv = xb[(size_t)nbr[k] * Din + d];
    A[k * As + d] = xiv;
    A[k * As + Din + d] = xjv - xiv;
  }
  __syncthreads();

  int hi = lane >> 4, l15 = lane & 15;
  const float* arow = A + l15 * As;
  for (int ct = wave; ct < nct; ct += 8) {
    v8f acc = {};
    int c = ct * 16 + l15;
#pragma unroll 8
    for (int kk = 0; kk < K2; kk += 4) {
      int k0 = kk + hi * 2;
      v2f a = *(const v2f*)(arow + k0);            // ds_load_b64
      v2f w;
      w.x = W[(size_t)k0 * Cout + c];              // coalesced b32 per K-row
      w.y = W[(size_t)(k0 + 1) * Cout + c];
      acc = wmma_f32(a, w, acc);
    }
    float bc = bias[c], gc = g[c], bec = be[c];
    float m = -INFINITY;
#pragma unroll
    for (int r = 0; r < 8; ++r) {
      float h = gc * ((acc[r] + bc) * kBN) + bec;
      m = fmaxf(m, fmaxf(h, 0.f));
    }
    m = fmaxf(m, __shfl_xor(m, 16, 32));  // rows 0-7 vs rows 8-15
    if (hi == 0) out[(size_t)p * Cout + c] = m;
  }
}

// ---------------------------------------------------------------- head
__global__ __launch_bounds__(256) void head_kernel(
    const float* __restrict__ h3, const float* __restrict__ Wf1,
    const float* __restrict__ bf1, const float* __restrict__ gf,
    const float* __restrict__ bef, const float* __restrict__ Wf2,
    const float* __restrict__ bf2, float* __restrict__ out) {
  int b = blockIdx.x, tid = threadIdx.x;
  __shared__ float pbuf[256];
  __shared__ float fbuf[128];
  const float* hb = h3 + (size_t)b * NN * 256;
  float m = -INFINITY;
  for (int nn2 = 0; nn2 < NN; ++nn2) m = fmaxf(m, hb[(size_t)nn2 * 256 + tid]);
  pbuf[tid] = m;
  __syncthreads();
  if (tid < 128) {
    float acc = bf1[tid];
#pragma unroll 8
    for (int k = 0; k < 256; ++k) acc += pbuf[k] * Wf1[k * 128 + tid];
    fbuf[tid] = fmaxf(gf[tid] * (acc * kBN) + bef[tid], 0.f);
  }
  __syncthreads();
  if (tid < 12) {
    float acc = bf2[tid];
#pragma unroll 8
    for (int k = 0; k < 128; ++k) acc += fbuf[k] * Wf2[k * 12 + tid];
    out[b * 12 + tid] = acc;
  }
}

// ---------------------------------------------------------------- launch
extern "C" void kernel_launch(void* const* d_in, const int* in_sizes, int n_in,
                              void* d_out, int out_size, void* d_ws, size_t ws_size,
                              hipStream_t stream) {
  const float* x   = (const float*)d_in[0];
  const float* W1  = (const float*)d_in[1];
  const float* b1  = (const float*)d_in[2];
  const float* g1  = (const float*)d_in[3];
  const float* be1 = (const float*)d_in[4];
  const float* W2  = (const float*)d_in[5];
  const float* b2  = (const float*)d_in[6];
  const float* g2  = (const float*)d_in[7];
  const float* be2 = (const float*)d_in[8];
  const float* W3  = (const float*)d_in[9];
  const float* b3  = (const float*)d_in[10];
  const float* g3  = (const float*)d_in[11];
  const float* be3 = (const float*)d_in[12];
  const float* Wf1 = (const float*)d_in[13];
  const float* bf1 = (const float*)d_in[14];
  const float* gf  = (const float*)d_in[15];
  const float* bef = (const float*)d_in[16];
  const float* Wf2 = (const float*)d_in[17];
  const float* bf2 = (const float*)d_in[18];
  float* out = (float*)d_out;

  char* ws = (char*)d_ws;
  size_t off = 0;
  float* sq   = (float*)(ws + off); off += (size_t)BB * NN * 4;
  int*   idx  = (int*)(ws + off);   off += (size_t)BB * NN * KNB * 4;
  float* xpad = (float*)(ws + off); off += (size_t)BB * NN * 8 * 4;
  float* h1   = (float*)(ws + off); off += (size_t)BB * NN * 64 * 4;
  float* h2   = (float*)(ws + off); off += (size_t)BB * NN * 128 * 4;
  float* h3   = (float*)(ws + off); off += (size_t)BB * NN * 256 * 4;

  dim3 gSq((BB * NN + 255) / 256), b256(256);
  dim3 gKnn(BB * 16);
  dim3 gPts(BB * NN);

  // layer 1 (6 -> 64): kNN runs on zero-padded Din=8 copy (Gram-identical)
  pad_kernel<<<gSq, b256, 0, stream>>>(x, xpad);
  sq_kernel<6><<<gSq, b256, 0, stream>>>(x, sq);
  knn_kernel<8><<<gKnn, b256, 0, stream>>>(xpad, sq, idx);
  econv_kernel<6, 64><<<gPts, b256, 0, stream>>>(x, idx, W1, b1, g1, be1, h1);
  // layer 2 (64 -> 128)
  sq_kernel<64><<<gSq, b256, 0, stream>>>(h1, sq);
  knn_kernel<64><<<gKnn, b256, 0, stream>>>(h1, sq, idx);
  econv_kernel<64, 128><<<gPts, b256, 0, stream>>>(h1, idx, W2, b2, g2, be2, h2);
  // layer 3 (128 -> 256)
  sq_kernel<128><<<gSq, b256, 0, stream>>>(h2, sq);
  knn_kernel<128><<<gKnn, b256, 0, stream>>>(h2, sq, idx);
  econv_kernel<128, 256><<<gPts, b256, 0, stream>>>(h2, idx, W3, b3, g3, be3, h3);
  // head
  head_kernel<<<dim3(BB), b256, 0, stream>>>(h3, Wf1, bf1, gf, bef, Wf2, bf2, out);
}